// NaiveQueryGuidedTokenSelector_2877628088837
// MI455X (gfx1250) — compile-verified
//
#include <hip/hip_runtime.h>
#include <hip/hip_bf16.h>
#include <stdint.h>

// Problem constants
#define BATCH    8
#define NTOK     6400      // 80*80
#define CIN      1024
#define QD       256
#define NKEEP    3200
#define HALFCNT  51200u    // threefry counter split point: (8*6400*2)/2

typedef float v2f __attribute__((ext_vector_type(2)));
typedef float v8f __attribute__((ext_vector_type(8)));

#if defined(__gfx1250__) && __has_builtin(__builtin_amdgcn_wmma_f32_16x16x4_f32)
#define USE_WMMA_F32 1
#endif

// ---------------------------------------------------------------------------
// Kernel A1: M_b[c,k] = scale * sum_q Q[b,q,c] * Wagg[q,k]   (256x2 per batch)
// One wave per 16x16 output tile (N padded 2->16), K=256 via v_wmma_f32_16x16x4_f32.
// B-operand is staged in LDS and fetched branchlessly (clamped col + 0/1 mult)
// so no EXEC divergence appears in the WMMA loop.
// ---------------------------------------------------------------------------
__global__ __launch_bounds__(32) void kprep_M(const float* __restrict__ Q,
                                              const float* __restrict__ Wagg,
                                              float* __restrict__ Mws) {
  __shared__ float sB[QD * 2];
  const int b    = blockIdx.y;
  const int c0   = blockIdx.x * 16;
  const int lane = threadIdx.x;
  const int half = lane >> 4;   // 0 | 1
  const int l16  = lane & 15;
  const float scale = 0.0625f;  // 256^-0.5
  const float* Qb = Q + (size_t)b * QD * QD;

  for (int idx = lane; idx < QD * 2; idx += 32) sB[idx] = Wagg[idx];
  __syncthreads();

  const float wsel = (l16 < 2) ? 1.0f : 0.0f;  // zero-pad cols 2..15
  const int   col  = (l16 < 2) ? l16 : 0;      // clamped (branchless load)

#ifdef USE_WMMA_F32
  v8f acc = {};
  #pragma unroll 4
  for (int k0 = 0; k0 < QD; k0 += 4) {
    const int ka = k0 + 2 * half;
    v2f a, bm;
    // A (16x4 f32): lane holds row m=l16; VGPR0 -> K=ka, VGPR1 -> K=ka+1
    a.x = Qb[(size_t)(ka + 0) * QD + (c0 + l16)];
    a.y = Qb[(size_t)(ka + 1) * QD + (c0 + l16)];
    // B (4x16 f32): lane holds col n=l16; rows ka / ka+1
    bm.x = sB[(ka + 0) * 2 + col] * wsel;
    bm.y = sB[(ka + 1) * 2 + col] * wsel;
    acc = __builtin_amdgcn_wmma_f32_16x16x4_f32(false, a, false, bm,
                                                (short)0, acc, false, false);
  }
  // D 16x16: VGPR r -> M = r + 8*half, N = l16
  if (l16 < 2) {
    #pragma unroll
    for (int r = 0; r < 8; ++r) {
      const int m = c0 + r + 8 * half;
      Mws[b * (QD * 2) + m * 2 + l16] = acc[r] * scale;
    }
  }
#else
  // scalar fallback (also what the host pass parses)
  if (l16 < 2) {
    for (int r = 0; r < 8; ++r) {
      const int m = c0 + r + 8 * half;
      float s = 0.0f;
      for (int q = 0; q < QD; ++q)
        s = fmaf(Qb[(size_t)q * QD + m], sB[q * 2 + l16], s);
      Mws[b * (QD * 2) + m * 2 + l16] = s * scale;
    }
  }
#endif
}

// ---------------------------------------------------------------------------
// Kernel A2: V_b[j,k] = sum_c Wproj[j,c] * M_b[c,k]   (1024x2 per batch)
// A-operand pairs are contiguous -> float2 (global_load_b64); B from LDS.
// ---------------------------------------------------------------------------
__global__ __launch_bounds__(32) void kprep_V(const float* __restrict__ Wproj,
                                              const float* __restrict__ Mws,
                                              float* __restrict__ Vws) {
  __shared__ float sB[QD * 2];
  const int b    = blockIdx.y;
  const int j0   = blockIdx.x * 16;
  const int lane = threadIdx.x;
  const int half = lane >> 4;
  const int l16  = lane & 15;
  const float* Mb = Mws + b * (QD * 2);

  for (int idx = lane; idx < QD * 2; idx += 32) sB[idx] = Mb[idx];
  __syncthreads();

  const float wsel = (l16 < 2) ? 1.0f : 0.0f;
  const int   col  = (l16 < 2) ? l16 : 0;

#ifdef USE_WMMA_F32
  v8f acc = {};
  #pragma unroll 4
  for (int k0 = 0; k0 < QD; k0 += 4) {
    const int ka = k0 + 2 * half;
    v2f a, bm;
    const float2 av =
        *(const float2*)&Wproj[(size_t)(j0 + l16) * QD + ka];  // b64 load
    a.x = av.x;
    a.y = av.y;
    bm.x = sB[(ka + 0) * 2 + col] * wsel;
    bm.y = sB[(ka + 1) * 2 + col] * wsel;
    acc = __builtin_amdgcn_wmma_f32_16x16x4_f32(false, a, false, bm,
                                                (short)0, acc, false, false);
  }
  if (l16 < 2) {
    #pragma unroll
    for (int r = 0; r < 8; ++r) {
      const int j = j0 + r + 8 * half;
      Vws[b * (CIN * 2) + j * 2 + l16] = acc[r];
    }
  }
#else
  if (l16 < 2) {
    for (int r = 0; r < 8; ++r) {
      const int j = j0 + r + 8 * half;
      float s = 0.0f;
      for (int c = 0; c < QD; ++c)
        s = fmaf(Wproj[(size_t)j * QD + c], sB[c * 2 + l16], s);
      Vws[b * (CIN * 2) + j * 2 + l16] = s;
    }
  }
#endif
}

// ---------------------------------------------------------------------------
// Kernel A3: c_b[k] = b_proj . M_b[:,k] + b_agg[k]
// ---------------------------------------------------------------------------
__global__ void kprep_c(const float* __restrict__ bproj,
                        const float* __restrict__ bagg,
                        const float* __restrict__ Mws,
                        float* __restrict__ cws) {
  const int b = blockIdx.x;
  const int n = threadIdx.x;  // 2 threads
  float s = 0.0f;
  for (int c = 0; c < QD; ++c)
    s = fmaf(bproj[c], Mws[b * (QD * 2) + c * 2 + n], s);
  cws[b * 2 + n] = s + bagg[n];
}

// ---------------------------------------------------------------------------
// Threefry-2x32-20, key = jax.random.key(42) -> (0, 42); exact JAX semantics.
// ---------------------------------------------------------------------------
__device__ __forceinline__ uint32_t rotl32(uint32_t v, int r) {
  return (v << r) | (v >> (32 - r));
}

__device__ __forceinline__ void threefry2x32(uint32_t x0, uint32_t x1,
                                             uint32_t& o0, uint32_t& o1) {
  const uint32_t ks0 = 0u, ks1 = 42u;
  const uint32_t ks2 = 0u ^ 42u ^ 0x1BD11BDAu;
  x0 += ks0; x1 += ks1;
#define TF_R(r) { x0 += x1; x1 = rotl32(x1, r); x1 ^= x0; }
  TF_R(13) TF_R(15) TF_R(26) TF_R(6)   x0 += ks1; x1 += ks2 + 1u;
  TF_R(17) TF_R(29) TF_R(16) TF_R(24)  x0 += ks2; x1 += ks0 + 2u;
  TF_R(13) TF_R(15) TF_R(26) TF_R(6)   x0 += ks0; x1 += ks1 + 3u;
  TF_R(17) TF_R(29) TF_R(16) TF_R(24)  x0 += ks1; x1 += ks2 + 4u;
  TF_R(13) TF_R(15) TF_R(26) TF_R(6)   x0 += ks2; x1 += ks0 + 5u;
#undef TF_R
  o0 = x0; o1 = x1;
}

__device__ __forceinline__ float gumbel_for(uint32_t f) {
  uint32_t bits, a, b2;
  if (f < HALFCNT) { threefry2x32(f, f + HALFCNT, a, b2); bits = a;  }
  else             { threefry2x32(f - HALFCNT, f, a, b2); bits = b2; }
  const float u01 = __uint_as_float((bits >> 9) | 0x3f800000u) - 1.0f;
  const float u   = 1e-10f + u01 * (1.0f - 1e-10f);
  return -logf(-logf(u));
}

// ---------------------------------------------------------------------------
// Kernel B: streaming fused pass. One wave per 4 consecutive tokens:
//   logits = mask * (x_row . V_b) + c_b ; log_softmax ; score ; gumbel mask.
// 210 MB read once -> memory bound (~9 us at 23.3 TB/s).
// ---------------------------------------------------------------------------
__global__ __launch_bounds__(256) void kmain(const float* __restrict__ x,
                                             const float* __restrict__ mask,
                                             const float* __restrict__ Vws,
                                             const float* __restrict__ cws,
                                             float* __restrict__ score_out,
                                             float* __restrict__ mask_out) {
  __shared__ float4 sV0[CIN / 4];
  __shared__ float4 sV1[CIN / 4];
  __shared__ float  sc[2];

  const int b   = blockIdx.y;
  const int tid = threadIdx.x;
  const float* Vb = Vws + b * (CIN * 2);
  for (int c = tid; c < CIN; c += 256) {
    ((float*)sV0)[c] = Vb[c * 2 + 0];
    ((float*)sV1)[c] = Vb[c * 2 + 1];
  }
  if (tid < 2) sc[tid] = cws[b * 2 + tid];
  __syncthreads();

  const int wave = tid >> 5;
  const int lane = tid & 31;
  const float c0 = sc[0], c1 = sc[1];

  #pragma unroll
  for (int t = 0; t < 4; ++t) {
    const int n = blockIdx.x * 32 + wave * 4 + t;
    const float4* __restrict__ xr =
        (const float4*)(x + ((size_t)b * NTOK + n) * CIN);

    if (t < 3) {  // prefetch next token row (speculative, 128B/lane)
      __builtin_prefetch((const char*)(xr + CIN / 4) + lane * 128, 0, 0);
    }

    float a0 = 0.0f, a1 = 0.0f;
    #pragma unroll
    for (int i = 0; i < 8; ++i) {
      const float4 xv = xr[i * 32 + lane];
      const float4 v0 = sV0[i * 32 + lane];
      const float4 v1 = sV1[i * 32 + lane];
      a0 = fmaf(xv.x, v0.x, a0); a0 = fmaf(xv.y, v0.y, a0);
      a0 = fmaf(xv.z, v0.z, a0); a0 = fmaf(xv.w, v0.w, a0);
      a1 = fmaf(xv.x, v1.x, a1); a1 = fmaf(xv.y, v1.y, a1);
      a1 = fmaf(xv.z, v1.z, a1); a1 = fmaf(xv.w, v1.w, a1);
    }
    // wave32 butterfly reduction
    #pragma unroll
    for (int off = 16; off >= 1; off >>= 1) {
      a0 += __shfl_xor(a0, off, 32);
      a1 += __shfl_xor(a1, off, 32);
    }

    if (lane == 0) {
      const float mk = mask[(size_t)b * NTOK + n];
      const float l0 = fmaf(mk, a0, c0);
      const float l1 = fmaf(mk, a1, c1);
      const float mx  = fmaxf(l0, l1);
      const float lse = mx + logf(expf(l0 - mx) + expf(l1 - mx));
      const float p0 = l0 - lse;
      const float p1 = l1 - lse;
      const uint32_t f0 = 2u * (uint32_t)(b * NTOK + n);
      const float g0 = gumbel_for(f0);
      const float g1 = gumbel_for(f0 + 1u);
      const float z  = (p1 + g1) - (p0 + g0);
      score_out[(size_t)b * NTOK + n] = p0;
      mask_out [(size_t)b * NTOK + n] = 1.0f / (1.0f + expf(z));
    }
  }
}

// ---------------------------------------------------------------------------
// Kernel C: stable descending argsort via rank counting (matches
// jnp.argsort(-score): score desc, ties by index asc). Ranks are unique ->
// conflict-free scatter. Inner scan reads LDS as float4 (ds_load_b128).
// ---------------------------------------------------------------------------
__global__ __launch_bounds__(128) void ksort(const float* __restrict__ score,
                                             float* __restrict__ keep_s,
                                             float* __restrict__ drop_s,
                                             int* __restrict__ keep_i,
                                             int* __restrict__ drop_i) {
  __shared__ float s[NTOK];
  const int b = blockIdx.y;
  const float* sb = score + (size_t)b * NTOK;
  for (int j = threadIdx.x; j < NTOK; j += 128) s[j] = sb[j];
  __syncthreads();

  const int i  = blockIdx.x * 128 + threadIdx.x;
  const float si = s[i];
  const float4* s4 = (const float4*)s;
  int rank = 0;
  #pragma unroll 4
  for (int j4 = 0; j4 < NTOK / 4; ++j4) {
    const float4 v = s4[j4];
    const int j = j4 * 4;
    rank += (v.x > si) | ((v.x == si) & (j + 0 < i));
    rank += (v.y > si) | ((v.y == si) & (j + 1 < i));
    rank += (v.z > si) | ((v.z == si) & (j + 2 < i));
    rank += (v.w > si) | ((v.w == si) & (j + 3 < i));
  }
  if (rank < NKEEP) {
    keep_s[b * NKEEP + rank] = si;
    keep_i[b * NKEEP + rank] = i;
  } else {
    drop_s[b * NKEEP + rank - NKEEP] = si;
    drop_i[b * NKEEP + rank - NKEEP] = i;
  }
}

// ---------------------------------------------------------------------------
extern "C" void kernel_launch(void* const* d_in, const int* in_sizes, int n_in,
                              void* d_out, int out_size, void* d_ws, size_t ws_size,
                              hipStream_t stream) {
  const float* x     = (const float*)d_in[0];  // (8,80,80,1024)
  const float* mask  = (const float*)d_in[1];  // (8,80,80,1)
  const float* Q     = (const float*)d_in[2];  // (8,256,256)
  const float* Wproj = (const float*)d_in[3];  // (1024,256)
  const float* bproj = (const float*)d_in[4];  // (256,)
  const float* Wagg  = (const float*)d_in[5];  // (256,2)
  const float* bagg  = (const float*)d_in[6];  // (2,)

  float* out = (float*)d_out;
  // Output layout (flat, return order):
  // [0)         keep_score  8*3200
  // [25600)     drop_score  8*3200
  // [51200)     keep_idx    8*3200 (int32 bits)
  // [76800)     drop_idx    8*3200 (int32 bits)
  // [102400)    new_mask    8*6400
  // [153600)    score       8*6400
  float* keep_s   = out;
  float* drop_s   = out + 25600;
  int*   keep_i   = (int*)d_out + 51200;
  int*   drop_i   = (int*)d_out + 76800;
  float* mask_out = out + 102400;
  float* score_o  = out + 153600;

  float* ws  = (float*)d_ws;
  float* Mws = ws;              // 8*256*2  = 4096 floats
  float* Vws = ws + 4096;       // 8*1024*2 = 16384 floats
  float* cws = ws + 20480;      // 8*2      = 16 floats

  dim3 gM(QD / 16, BATCH);      // 16 x 8 waves
  kprep_M<<<gM, 32, 0, stream>>>(Q, Wagg, Mws);

  dim3 gV(CIN / 16, BATCH);     // 64 x 8 waves
  kprep_V<<<gV, 32, 0, stream>>>(Wproj, Mws, Vws);

  kprep_c<<<BATCH, 2, 0, stream>>>(bproj, bagg, Mws, cws);

  dim3 gB(NTOK / 32, BATCH);    // 200 x 8 blocks of 256 threads (8 waves)
  kmain<<<gB, 256, 0, stream>>>(x, mask, Vws, cws, score_o, mask_out);

  dim3 gC(NTOK / 128, BATCH);   // 50 x 8 blocks
  ksort<<<gC, 128, 0, stream>>>(score_o, keep_s, drop_s, keep_i, drop_i);
}